// NeuralMemory_6897717477466
// MI455X (gfx1250) — compile-verified
//
#include <hip/hip_runtime.h>
#include <math.h>

#define Hd 128
#define H2 256
#define Sd 256
#define Bd 4
#define NT 512

// momentum layout per batch (floats)
#define OFF_W1_0 0
#define OFF_B1_0 32768
#define OFF_W2_0 33024
#define OFF_B2_0 65792
#define OFF_W1_1 65920
#define OFF_B1_1 98688
#define OFF_W2_1 98944
#define OFF_B2_1 131712
#define PB       131840

typedef __attribute__((ext_vector_type(2))) float v2f;
typedef __attribute__((ext_vector_type(8))) float v8f;

__device__ __forceinline__ float sigmoidf_(float x) { return 1.f / (1.f + __expf(-x)); }
__device__ __forceinline__ float siluf_(float x)    { return x * sigmoidf_(x); }
__device__ __forceinline__ float dsiluf_(float z)   { float s = sigmoidf_(z); return s * (1.f + z * (1.f - s)); }

// ---------------------------------------------------------------------------
// Kernel 1: q/k/v projections + gates via fp32 WMMA (v_wmma_f32_16x16x4_f32)
// x: [1024,128], W: [128,128]. 64 blocks x 256 threads; wave w owns N-tile w.
// Software-pipelined: preload all A fragments once, per matrix preload all
// 32 B fragments (clustered loads, single wait), then 32 back-to-back WMMAs.
// ---------------------------------------------------------------------------
__global__ __launch_bounds__(256)
void proj_kernel(const float* __restrict__ x,
                 const float* __restrict__ Wq, const float* __restrict__ Wk, const float* __restrict__ Wv,
                 const float* __restrict__ w_lr, const float* __restrict__ b_lr,
                 const float* __restrict__ w_fg, const float* __restrict__ b_fg,
                 const float* __restrict__ w_mo, const float* __restrict__ b_mo,
                 float* __restrict__ qn, float* __restrict__ kn, float* __restrict__ vb,
                 float* __restrict__ th, float* __restrict__ al, float* __restrict__ et)
{
    __shared__ float tile[3][16 * Hd];
    __shared__ float red[256];
    __shared__ float norms[2][16];

    const int tid  = threadIdx.x;
    const int lane = tid & 31;
    const int wave = tid >> 5;          // 0..7 -> N-tile
    const int m0   = blockIdx.x * 16;   // row base in [0,1024)
    const int n0   = wave * 16;
    const int lr   = lane & 15;
    const int hi   = lane >> 4;         // 0: K={0,1}, 1: K={2,3} halves
    const int arow = m0 + lr;

    // Preload all 32 A fragments (K chunks of 4); contiguous pair -> b64 load.
    v2f afr[32];
#pragma unroll
    for (int c = 0; c < 32; ++c) {
        const float* ap = x + arow * Hd + 4 * c + 2 * hi;
        v2f a; a.x = ap[0]; a.y = ap[1];
        afr[c] = a;
    }

    const float* Ws[3] = { Wq, Wk, Wv };
    for (int mtx = 0; mtx < 3; ++mtx) {
        const float* W = Ws[mtx];
        // Preload all B fragments for this matrix's N-tile.
        v2f bfr[32];
#pragma unroll
        for (int c = 0; c < 32; ++c) {
            const float* bp = W + (4 * c + 2 * hi) * Hd + n0 + lr;
            v2f bv; bv.x = bp[0]; bv.y = bp[Hd];
            bfr[c] = bv;
        }
        v8f cacc = { 0.f, 0.f, 0.f, 0.f, 0.f, 0.f, 0.f, 0.f };
#pragma unroll
        for (int c = 0; c < 32; ++c)
            cacc = __builtin_amdgcn_wmma_f32_16x16x4_f32(false, afr[c], false, bfr[c],
                                                         (short)0, cacc, false, false);
#pragma unroll
        for (int r = 0; r < 8; ++r) {
            int row = r + 8 * hi;       // C/D layout: vgpr r -> M=r (lanes<16) / M=r+8
            tile[mtx][row * Hd + n0 + lr] = cacc[r];
        }
    }
    __syncthreads();

    // silu in place
    for (int e = 0; e < 8; ++e) {
        int idx = e * 256 + tid;
        tile[0][idx] = siluf_(tile[0][idx]);
        tile[1][idx] = siluf_(tile[1][idx]);
        tile[2][idx] = siluf_(tile[2][idx]);
    }
    __syncthreads();

    // row L2 norms for q and k
    for (int mtx = 0; mtx < 2; ++mtx) {
        int row = tid >> 4, seg = tid & 15;
        float p = 0.f;
        for (int cc = 0; cc < 8; ++cc) {
            float v = tile[mtx][row * Hd + seg * 8 + cc];
            p += v * v;
        }
        red[tid] = p;
        __syncthreads();
        if (tid < 16) {
            float s = 0.f;
            for (int g = 0; g < 16; ++g) s += red[tid * 16 + g];
            norms[mtx][tid] = fmaxf(sqrtf(s), 1e-12f);
        }
        __syncthreads();
    }

    // write q (normalized), k (normalized), v
    for (int e = 0; e < 8; ++e) {
        int idx = e * 256 + tid;
        int row = idx >> 7, col = idx & 127;
        int grow = m0 + row;
        qn[grow * Hd + col] = tile[0][idx] / norms[0][row];
        kn[grow * Hd + col] = tile[1][idx] / norms[1][row];
        vb[grow * Hd + col] = tile[2][idx];
    }

    // gates: theta/alpha/eta per row
    if (tid < 16) {
        int grow = m0 + tid;
        float a0 = 0.f, a1 = 0.f, a2 = 0.f;
        for (int c = 0; c < Hd; ++c) {
            float xv = x[grow * Hd + c];
            a0 += xv * w_lr[c];
            a1 += xv * w_fg[c];
            a2 += xv * w_mo[c];
        }
        th[grow] = 0.01f * sigmoidf_(a0 + b_lr[0]);
        al[grow] = sigmoidf_(a1 + b_fg[0]);
        et[grow] = sigmoidf_(a2 + b_mo[0]);
    }
}

// ---------------------------------------------------------------------------
// Kernel 2: broadcast initial weight momentum into per-batch workspace
// ---------------------------------------------------------------------------
__global__ void minit_kernel(const float* __restrict__ mw10, const float* __restrict__ mb10,
                             const float* __restrict__ mw20, const float* __restrict__ mb20,
                             const float* __restrict__ mw11, const float* __restrict__ mb11,
                             const float* __restrict__ mw21, const float* __restrict__ mb21,
                             float* __restrict__ m)
{
    int idx = blockIdx.x * blockDim.x + threadIdx.x;
    if (idx >= Bd * PB) return;
    int r = idx % PB;
    float v;
    if      (r < OFF_B1_0) v = mw10[r];
    else if (r < OFF_W2_0) v = mb10[r - OFF_B1_0];
    else if (r < OFF_B2_0) v = mw20[r - OFF_W2_0];
    else if (r < OFF_W1_1) v = mb20[r - OFF_B2_0];
    else if (r < OFF_B1_1) v = mw11[r - OFF_W1_1];
    else if (r < OFF_W2_1) v = mb11[r - OFF_B1_1];
    else if (r < OFF_B2_1) v = mw21[r - OFF_W2_1];
    else                   v = mb21[r - OFF_B2_1];
    m[idx] = v;
}

// ---------------------------------------------------------------------------
// Scan helpers: weights materialized on the fly as p0*decay + ms*m
// ---------------------------------------------------------------------------
__device__ __forceinline__
void matvec_fwd(const float* __restrict__ p0, const float* __restrict__ mm,
                const float* __restrict__ bp0, const float* __restrict__ mb,
                const float* xin, const float* res, float* zout,
                int R, int C, float decay, float ms, float* red, int tid)
{
    int gpc = NT / C, rpg = R / gpc;
    int j = tid % C, g = tid / C;
    float acc = 0.f;
    for (int r = 0; r < rpg; ++r) {
        int i = g * rpg + r;
        float w = p0[i * C + j] * decay + ms * mm[i * C + j];
        acc += xin[i] * w;
    }
    red[tid] = acc;
    __syncthreads();
    if (tid < C) {
        float s = bp0[j] * decay + ms * mb[j];
        for (int g2 = 0; g2 < gpc; ++g2) s += red[g2 * C + j];
        if (res) s += res[j];
        zout[j] = s;
    }
    __syncthreads();
}

__device__ __forceinline__
void matvecT(const float* __restrict__ p0, const float* __restrict__ mm,
             const float* dvec, const float* res, float* outv,
             int R, int C, float decay, float ms, float* red, int tid)
{
    int gpo = NT / R, cpg = C / gpo;
    int i = tid % R, g = tid / R;
    float acc = 0.f;
    for (int jj = 0; jj < cpg; ++jj) {
        int j = g * cpg + jj;
        acc += dvec[j] * (p0[i * C + j] * decay + ms * mm[i * C + j]);
    }
    red[tid] = acc;
    __syncthreads();
    if (tid < R) {
        float s = res ? res[tid] : 0.f;
        for (int g2 = 0; g2 < gpo; ++g2) s += red[g2 * R + tid];
        outv[tid] = s;
    }
    __syncthreads();
}

// Momentum update fused with the forward matvecs for query (this step) and
// the NEXT step's key, so each weight element is touched exactly once.
__device__ __forceinline__
void update_fused(const float* __restrict__ p0, float* __restrict__ mm,
                  const float* __restrict__ bp0, float* __restrict__ mb,
                  const float* gr, const float* gc,
                  const float* xq, const float* xk,
                  const float* resq, const float* resk,
                  float* zq, float* zk,
                  int R, int C, float eta, float theta, float decay,
                  float* red, float* red2, int tid)
{
    int gpc = NT / C, rpg = R / gpc;
    int j = tid % C, g = tid / C;
    float gcj = gc[j];
    float accq = 0.f, acck = 0.f;
    for (int r = 0; r < rpg; ++r) {
        int i = g * rpg + r;
        int idx = i * C + j;
        float mv = eta * mm[idx] - theta * (gr[i] * gcj);
        mm[idx] = mv;
        float w = p0[idx] * decay + mv;
        accq += xq[i] * w;
        acck += xk[i] * w;
    }
    red[tid] = accq;
    red2[tid] = acck;
    __syncthreads();
    if (tid < C) {
        float mbv = eta * mb[j] - theta * gcj;
        mb[j] = mbv;
        float be = bp0[j] * decay + mbv;
        float sq = be, sk = be;
        for (int g2 = 0; g2 < gpc; ++g2) { sq += red[g2 * C + j]; sk += red2[g2 * C + j]; }
        if (resq) { sq += resq[j]; sk += resk[j]; }
        zq[j] = sq;
        zk[j] = sk;
    }
    __syncthreads();
}

// act buffer layout (floats): kx 0, z0 128, s0 384, h1 640, z1 768, s1 1024, pred 1280
__global__ __launch_bounds__(NT)
void scan_kernel(const float* __restrict__ qbuf, const float* __restrict__ kbuf, const float* __restrict__ vbuf,
                 const float* __restrict__ th, const float* __restrict__ al, const float* __restrict__ et,
                 const float* __restrict__ w10, const float* __restrict__ b10,
                 const float* __restrict__ w20, const float* __restrict__ b20,
                 const float* __restrict__ w11, const float* __restrict__ b11,
                 const float* __restrict__ w21, const float* __restrict__ b21,
                 const float* __restrict__ mb10, const float* __restrict__ mb20,
                 const float* __restrict__ mb11, const float* __restrict__ mb21,
                 float* __restrict__ mws, float* __restrict__ out)
{
    const int b = blockIdx.x;
    const int tid = threadIdx.x;
    float* m = mws + b * PB;
    float* mW1[2]       = { m + OFF_W1_0, m + OFF_W1_1 };
    float* mW2[2]       = { m + OFF_W2_0, m + OFF_W2_1 };
    const float* pW1[2] = { w10, w11 };
    const float* pW2[2] = { w20, w21 };
    const float* pB1[2] = { b10, b11 };
    const float* pB2[2] = { b20, b21 };

    __shared__ float mb1s[2][H2];
    __shared__ float mb2s[2][Hd];
    __shared__ float kact[2][1408];
    __shared__ float qact[1408];
    __shared__ float dd[768];     // d2 0, dz1 128, dh1 384, dz0 512
    __shared__ float red[NT], red2[NT];

    if (tid < H2) { mb1s[0][tid] = mb10[tid]; mb1s[1][tid] = mb11[tid]; }
    if (tid < Hd) { mb2s[0][tid] = mb20[tid]; mb2s[1][tid] = mb21[tid]; }
    if (tid < Hd) kact[0][tid] = kbuf[(b * Sd + 0) * Hd + tid];
    __syncthreads();

    // forward(k_0) with carried params == p0 (decay=1, momentum scale=0)
    int cur = 0;
    float* KC = kact[0];
    matvec_fwd(pW1[0], mW1[0], pB1[0], mb1s[0], KC + 0,    nullptr,  KC + 128,  128, 256, 1.f, 0.f, red, tid);
    if (tid < H2) KC[384 + tid] = siluf_(KC[128 + tid]);
    __syncthreads();
    matvec_fwd(pW2[0], mW2[0], pB2[0], mb2s[0], KC + 384,  KC + 0,   KC + 640,  256, 128, 1.f, 0.f, red, tid);
    matvec_fwd(pW1[1], mW1[1], pB1[1], mb1s[1], KC + 640,  nullptr,  KC + 768,  128, 256, 1.f, 0.f, red, tid);
    if (tid < H2) KC[1024 + tid] = siluf_(KC[768 + tid]);
    __syncthreads();
    matvec_fwd(pW2[1], mW2[1], pB2[1], mb2s[1], KC + 1024, KC + 640, KC + 1280, 256, 128, 1.f, 0.f, red, tid);

    float decay_prev = 1.f, ms_prev = 0.f;
    for (int t = 0; t < Sd; ++t) {
        const int idx = b * Sd + t;
        const float theta_t = th[idx], alpha_t = al[idx], eta_t = et[idx];
        const float* vrow = vbuf + idx * Hd;
        const float* qrow = qbuf + idx * Hd;
        const int tn = (t + 1 < Sd) ? (t + 1) : (Sd - 1);
        const float* krow_n = kbuf + (b * Sd + tn) * Hd;
        KC = kact[cur];
        float* KN = kact[1 - cur];

        __builtin_prefetch(qrow, 0, 0);     // -> global_prefetch_b8
        __builtin_prefetch(krow_n, 0, 0);

        // ---- backward through p_prev = p0*decay_prev + ms_prev*m ----
        if (tid < Hd) dd[tid] = (KC[1280 + tid] - vrow[tid]) * (2.f / (float)Hd);
        __syncthreads();
        matvecT(pW2[1], mW2[1], dd + 0,   nullptr, dd + 128, 256, 128, decay_prev, ms_prev, red, tid); // ds1
        if (tid < H2) dd[128 + tid] *= dsiluf_(KC[768 + tid]);                                          // dz1
        __syncthreads();
        matvecT(pW1[1], mW1[1], dd + 128, dd + 0,  dd + 384, 128, 256, decay_prev, ms_prev, red, tid); // dh1
        matvecT(pW2[0], mW2[0], dd + 384, nullptr, dd + 512, 256, 128, decay_prev, ms_prev, red, tid); // ds0
        if (tid < H2) dd[512 + tid] *= dsiluf_(KC[128 + tid]);                                          // dz0
        if (tid < Hd) { qact[tid] = qrow[tid]; KN[tid] = krow_n[tid]; }
        __syncthreads();

        // ---- momentum update + fused forward for q_t and k_{t+1} ----
        const float dec = 1.f - alpha_t;
        update_fused(pW1[0], mW1[0], pB1[0], mb1s[0], KC + 0,    dd + 512, qact + 0,    KN + 0,
                     nullptr,    nullptr,  qact + 128,  KN + 128,  128, 256, eta_t, theta_t, dec, red, red2, tid);
        if (tid < H2) { qact[384 + tid] = siluf_(qact[128 + tid]); KN[384 + tid] = siluf_(KN[128 + tid]); }
        __syncthreads();
        update_fused(pW2[0], mW2[0], pB2[0], mb2s[0], KC + 384,  dd + 384, qact + 384,  KN + 384,
                     qact + 0,   KN + 0,   qact + 640,  KN + 640,  256, 128, eta_t, theta_t, dec, red, red2, tid);
        update_fused(pW1[1], mW1[1], pB1[1], mb1s[1], KC + 640,  dd + 128, qact + 640,  KN + 640,
                     nullptr,    nullptr,  qact + 768,  KN + 768,  128, 256, eta_t, theta_t, dec, red, red2, tid);
        if (tid < H2) { qact[1024 + tid] = siluf_(qact[768 + tid]); KN[1024 + tid] = siluf_(KN[768 + tid]); }
        __syncthreads();
        update_fused(pW2[1], mW2[1], pB2[1], mb2s[1], KC + 1024, dd + 0,   qact + 1024, KN + 1024,
                     qact + 640, KN + 640, qact + 1280, KN + 1280, 256, 128, eta_t, theta_t, dec, red, red2, tid);

        if (tid < Hd) out[idx * Hd + tid] = qact[1280 + tid];

        decay_prev = dec;
        ms_prev = 1.f;
        cur ^= 1;
        __syncthreads();
    }
}

// ---------------------------------------------------------------------------
extern "C" void kernel_launch(void* const* d_in, const int* in_sizes, int n_in,
                              void* d_out, int out_size, void* d_ws, size_t ws_size,
                              hipStream_t stream)
{
    const float* x    = (const float*)d_in[0];
    const float* Wq   = (const float*)d_in[1];
    const float* Wk   = (const float*)d_in[2];
    const float* Wv   = (const float*)d_in[3];
    const float* w_lr = (const float*)d_in[4];
    const float* b_lr = (const float*)d_in[5];
    const float* w_fg = (const float*)d_in[6];
    const float* b_fg = (const float*)d_in[7];
    const float* w_mo = (const float*)d_in[8];
    const float* b_mo = (const float*)d_in[9];
    const float* w10  = (const float*)d_in[10];
    const float* b10  = (const float*)d_in[11];
    const float* w20  = (const float*)d_in[12];
    const float* b20  = (const float*)d_in[13];
    const float* mw10 = (const float*)d_in[14];
    const float* mb10 = (const float*)d_in[15];
    const float* mw20 = (const float*)d_in[16];
    const float* mb20 = (const float*)d_in[17];
    const float* w11  = (const float*)d_in[18];
    const float* b11  = (const float*)d_in[19];
    const float* w21  = (const float*)d_in[20];
    const float* b21  = (const float*)d_in[21];
    const float* mw11 = (const float*)d_in[22];
    const float* mb11 = (const float*)d_in[23];
    const float* mw21 = (const float*)d_in[24];
    const float* mb21 = (const float*)d_in[25];

    float* ws  = (float*)d_ws;
    float* qn  = ws;                // 131072
    float* kn  = qn + 131072;       // 131072
    float* vb  = kn + 131072;       // 131072
    float* th  = vb + 131072;       // 1024
    float* al  = th + 1024;         // 1024
    float* et  = al + 1024;         // 1024
    float* mws = et + 1024;         // Bd*PB = 527360

    proj_kernel<<<64, 256, 0, stream>>>(x, Wq, Wk, Wv, w_lr, b_lr, w_fg, b_fg, w_mo, b_mo,
                                        qn, kn, vb, th, al, et);
    minit_kernel<<<(Bd * PB + 255) / 256, 256, 0, stream>>>(mw10, mb10, mw20, mb20,
                                                            mw11, mb11, mw21, mb21, mws);
    scan_kernel<<<Bd, NT, 0, stream>>>(qn, kn, vb, th, al, et,
                                       w10, b10, w20, b20, w11, b11, w21, b21,
                                       mb10, mb20, mb11, mb21,
                                       mws, (float*)d_out);
}